// DLightGCL_6313601925361
// MI455X (gfx1250) — compile-verified
//
#include <hip/hip_runtime.h>
#include <math.h>

#define N_USERS 50000
#define N_ITEMS 50000
#define DIM     128
#define QD      64
#define NE      1000000
#define NB      1024
#define TEMPF   0.2f
#define BETAF   0.1f

typedef float v2f __attribute__((ext_vector_type(2)));
typedef float v8f __attribute__((ext_vector_type(8)));

__device__ __forceinline__ v8f wmma4(v2f a, v2f b, v8f c) {
  // V_WMMA_F32_16X16X4_F32 : D = A(16x4) * B(4x16) + C
  return __builtin_amdgcn_wmma_f32_16x16x4_f32(false, a, false, b, (short)0, c,
                                               false, false);
}

__device__ __forceinline__ void atomAddF(float* p, float v) {
  __hip_atomic_fetch_add(p, v, __ATOMIC_RELAXED, __HIP_MEMORY_SCOPE_AGENT);
}

__device__ __forceinline__ float waveSum(float s) {
#pragma unroll
  for (int m = 16; m >= 1; m >>= 1) s += __shfl_xor(s, m, 32);
  return s;
}

// ---------------- row normalization: Xn = X / max(||row||, 1e-8) -----------
__global__ void normalize_rows_kernel(const float* __restrict__ X,
                                      float* __restrict__ Xn, int nrows) {
  int w = (blockIdx.x * blockDim.x + threadIdx.x) >> 5;
  int lane = threadIdx.x & 31;
  if (w >= nrows) return;
  const float4 v = ((const float4*)(X + (size_t)w * DIM))[lane];
  float s = v.x * v.x + v.y * v.y + v.z * v.z + v.w * v.w;
  s = waveSum(s);
  float inv = 1.0f / fmaxf(sqrtf(s), 1e-8f);
  float4 o;
  o.x = v.x * inv; o.y = v.y * inv; o.z = v.z * inv; o.w = v.w * inv;
  ((float4*)(Xn + (size_t)w * DIM))[lane] = o;
}

// ---------------- scatter H: user_H[r] += w*inr[c]; item_H[c] += w*un[r] ----
__global__ void scatter_H_kernel(const float* __restrict__ un,
                                 const float* __restrict__ inr,
                                 const int* __restrict__ rows,
                                 const int* __restrict__ cols,
                                 const float* __restrict__ av,
                                 float* __restrict__ uH, float* __restrict__ iH) {
  int e = (blockIdx.x * blockDim.x + threadIdx.x) >> 5;
  int lane = threadIdx.x & 31;
  if (e >= NE) return;
  int r = rows[e], c = cols[e];
  float wv = av[e];
  int off = lane * 4;
  const float4 vi = *(const float4*)(inr + (size_t)c * DIM + off);
  float* du = uH + (size_t)r * DIM + off;
  atomAddF(du + 0, wv * vi.x); atomAddF(du + 1, wv * vi.y);
  atomAddF(du + 2, wv * vi.z); atomAddF(du + 3, wv * vi.w);
  const float4 vu = *(const float4*)(un + (size_t)r * DIM + off);
  float* di = iH + (size_t)c * DIM + off;
  atomAddF(di + 0, wv * vu.x); atomAddF(di + 1, wv * vu.y);
  atomAddF(di + 2, wv * vu.z); atomAddF(di + 3, wv * vu.w);
}

// ---------------- per-edge cosine sim + row/col degree accumulation --------
__global__ void edge_cos_kernel(const float* __restrict__ uH,
                                const float* __restrict__ iH,
                                const int* __restrict__ rows,
                                const int* __restrict__ cols,
                                float* __restrict__ sim, float* __restrict__ rd,
                                float* __restrict__ cd) {
  int e = (blockIdx.x * blockDim.x + threadIdx.x) >> 5;
  int lane = threadIdx.x & 31;
  if (e >= NE) return;
  int r = rows[e], c = cols[e];
  int off = lane * 4;
  const float4 a = *(const float4*)(uH + (size_t)r * DIM + off);
  const float4 b = *(const float4*)(iH + (size_t)c * DIM + off);
  float dp = a.x * b.x + a.y * b.y + a.z * b.z + a.w * b.w;
  float na = a.x * a.x + a.y * a.y + a.z * a.z + a.w * a.w;
  float nb = b.x * b.x + b.y * b.y + b.z * b.z + b.w * b.w;
  dp = waveSum(dp); na = waveSum(na); nb = waveSum(nb);
  if (lane == 0) {
    float an = fmaxf(sqrtf(na), 1e-8f);
    float bn = fmaxf(sqrtf(nb), 1e-8f);
    float cs = dp / (an * bn);
    float sm = (cs + 1.0f) * 0.5f;
    sm = (sm > BETAF) ? sm : 0.0f;
    sim[e] = sm;
    if (sm > 0.0f) { atomAddF(&rd[r], sm); atomAddF(&cd[c], sm); }
  }
}

// ---------------- per-edge normalized adjacency value ----------------------
__global__ void edge_vals_kernel(const float* __restrict__ sim,
                                 const int* __restrict__ rows,
                                 const int* __restrict__ cols,
                                 const float* __restrict__ rd,
                                 const float* __restrict__ cd,
                                 float* __restrict__ vals) {
  int e = blockIdx.x * blockDim.x + threadIdx.x;
  if (e >= NE) return;
  float s = sim[e];
  float v = 0.0f;
  if (s > 0.0f) {
    float den = fmaxf(rd[rows[e]] * cd[cols[e]], 1e-12f);
    v = s * rsqrtf(den);
  }
  vals[e] = v;
}

// ---------------- SpMM: Z_u[r] += v*ei[c]; Z_i[c] += v*eu[r] ---------------
__global__ void scatter_Z_kernel(const float* __restrict__ eu,
                                 const float* __restrict__ ei,
                                 const int* __restrict__ rows,
                                 const int* __restrict__ cols,
                                 const float* __restrict__ vals,
                                 float* __restrict__ Zu, float* __restrict__ Zi) {
  int e = (blockIdx.x * blockDim.x + threadIdx.x) >> 5;
  int lane = threadIdx.x & 31;
  if (e >= NE) return;
  float v = vals[e];
  if (v == 0.0f) return;
  int r = rows[e], c = cols[e];
  int off = lane * 4;
  const float4 vi = *(const float4*)(ei + (size_t)c * DIM + off);
  float* du = Zu + (size_t)r * DIM + off;
  atomAddF(du + 0, v * vi.x); atomAddF(du + 1, v * vi.y);
  atomAddF(du + 2, v * vi.z); atomAddF(du + 3, v * vi.w);
  const float4 vu = *(const float4*)(eu + (size_t)r * DIM + off);
  float* di = Zi + (size_t)c * DIM + off;
  atomAddF(di + 0, v * vu.x); atomAddF(di + 1, v * vu.y);
  atomAddF(di + 2, v * vu.z); atomAddF(di + 3, v * vu.w);
}

// ------- T += A[QD x K] @ B[K x DIM]  (split-K, WMMA f32 16x16x4) ----------
__global__ void gemm_qd_splitk_kernel(const float* __restrict__ A,
                                      const float* __restrict__ Bm,
                                      float* __restrict__ T, int K, int kchunk) {
  int w = (blockIdx.x * blockDim.x + threadIdx.x) >> 5;
  int lane = threadIdx.x & 31, hlf = lane >> 4, l16 = lane & 15;
  int tile = w & 31, split = w >> 5;   // 32 tiles: 4 (M) x 8 (N)
  int m0 = (tile >> 3) << 4;
  int n0 = (tile & 7) << 4;
  int k0 = split * kchunk;
  int k1 = k0 + kchunk; if (k1 > K) k1 = K;
  v8f acc = {0.f, 0.f, 0.f, 0.f, 0.f, 0.f, 0.f, 0.f};
  const float* Arow = A + (size_t)(m0 + l16) * K;
  for (int k = k0; k < k1; k += 4) {
    v2f a; a.x = Arow[k + 2 * hlf]; a.y = Arow[k + 2 * hlf + 1];
    const float* Bp = Bm + (size_t)(k + 2 * hlf) * DIM + n0 + l16;
    v2f b; b.x = Bp[0]; b.y = Bp[DIM];
    acc = wmma4(a, b, acc);
  }
#pragma unroll
  for (int r = 0; r < 8; r++)
    atomAddF(&T[(m0 + r + 8 * hlf) * DIM + n0 + l16], acc[r]);
}

// ------- G = C0 + Am[M x QD] @ T[QD x DIM]  (WMMA) -------------------------
__global__ void gemm_md_kernel(const float* __restrict__ Am,
                               const float* __restrict__ T,
                               const float* __restrict__ C0,
                               float* __restrict__ G, int M) {
  int w = (blockIdx.x * blockDim.x + threadIdx.x) >> 5;
  int lane = threadIdx.x & 31, hlf = lane >> 4, l16 = lane & 15;
  int mt = w >> 3, nt = w & 7;
  int m0 = mt * 16, n0 = nt * 16;
  if (m0 >= M) return;
  v8f acc;
#pragma unroll
  for (int r = 0; r < 8; r++)
    acc[r] = C0[(size_t)(m0 + r + 8 * hlf) * DIM + n0 + l16];
  const float* Arow = Am + (size_t)(m0 + l16) * QD;
#pragma unroll
  for (int k = 0; k < QD; k += 4) {
    v2f a; a.x = Arow[k + 2 * hlf]; a.y = Arow[k + 2 * hlf + 1];
    const float* Bp = T + (size_t)(k + 2 * hlf) * DIM + n0 + l16;
    v2f b; b.x = Bp[0]; b.y = Bp[DIM];
    acc = wmma4(a, b, acc);
  }
#pragma unroll
  for (int r = 0; r < 8; r++)
    G[(size_t)(m0 + r + 8 * hlf) * DIM + n0 + l16] = acc[r];
}

// ------- streaming InfoNCE negatives: rowsum[b] += sum_j exp(s*A.bank^T) ---
__global__ void infonce_neg_kernel(const float* __restrict__ A,
                                   const float* __restrict__ bank, float scale,
                                   float* __restrict__ rowsum) {
  int w = (blockIdx.x * blockDim.x + threadIdx.x) >> 5;
  int lane = threadIdx.x & 31, hlf = lane >> 4, l16 = lane & 15;
  int mt = w & 63;      // 1024/16 anchor tiles
  int nt = w >> 6;      // 50000/16 bank tiles
  int m0 = mt * 16, j0 = nt * 16;
  v8f acc = {0.f, 0.f, 0.f, 0.f, 0.f, 0.f, 0.f, 0.f};
  const float* Arow = A + (size_t)(m0 + l16) * DIM;
  const float* Brow = bank + (size_t)(j0 + l16) * DIM;  // B[k][n] = bank[j0+n][k]
#pragma unroll
  for (int k = 0; k < DIM; k += 4) {
    v2f a; a.x = Arow[k + 2 * hlf]; a.y = Arow[k + 2 * hlf + 1];
    v2f b; b.x = Brow[k + 2 * hlf]; b.y = Brow[k + 2 * hlf + 1];
    acc = wmma4(a, b, acc);
  }
  float ex[8];
#pragma unroll
  for (int r = 0; r < 8; r++) ex[r] = __expf(acc[r] * scale);
#pragma unroll
  for (int m = 1; m < 16; m <<= 1) {
#pragma unroll
    for (int r = 0; r < 8; r++) ex[r] += __shfl_xor(ex[r], m, 32);
  }
  if (l16 == 0) {
#pragma unroll
    for (int r = 0; r < 8; r++) atomAddF(&rowsum[m0 + r + 8 * hlf], ex[r]);
  }
}

// ---------------- elementwise: o = a + b + c -------------------------------
__global__ void add3_kernel(const float* __restrict__ a,
                            const float* __restrict__ b,
                            const float* __restrict__ c, float* __restrict__ o,
                            int n) {
  int i = blockIdx.x * blockDim.x + threadIdx.x;
  if (i < n) o[i] = a[i] + b[i] + c[i];
}

// ---------------- sum of squares reduction ---------------------------------
__global__ void sumsq_kernel(const float* __restrict__ x, int n,
                             float* __restrict__ out) {
  float s = 0.0f;
  for (int i = blockIdx.x * blockDim.x + threadIdx.x; i < n;
       i += gridDim.x * blockDim.x) {
    float v = x[i];
    s += v * v;
  }
  s = waveSum(s);
  if ((threadIdx.x & 31) == 0) atomAddF(out, s);
}

// ---------------- gather rows ----------------------------------------------
__global__ void gather_rows_kernel(const float* __restrict__ src,
                                   const int* __restrict__ ids,
                                   float* __restrict__ dst) {
  int b = blockIdx.x, t = threadIdx.x;
  dst[(size_t)b * DIM + t] = src[(size_t)ids[b] * DIM + t];
}

// ---------------- final loss assembly --------------------------------------
__device__ __forceinline__ float dot128(const float* __restrict__ x,
                                        const float* __restrict__ y) {
  const float4* a = (const float4*)x;
  const float4* b = (const float4*)y;
  float s = 0.0f;
#pragma unroll
  for (int i = 0; i < 32; i++) {
    float4 u = a[i], v = b[i];
    s += u.x * v.x + u.y * v.y + u.z * v.z + u.w * v.w;
  }
  return s;
}

__device__ __forceinline__ float clamp5(float x) {
  return fminf(fmaxf(x, -5.0f), 5.0f);
}

__device__ float blockSum1024(float v, float* red) {
  int t = threadIdx.x;
  red[t] = v;
  __syncthreads();
  for (int s = 512; s > 0; s >>= 1) {
    if (t < s) red[t] += red[t + s];
    __syncthreads();
  }
  float r = red[0];
  __syncthreads();
  return r;
}

__global__ __launch_bounds__(1024) void final_kernel(
    const float* __restrict__ Gu, const float* __restrict__ Gi,
    const float* __restrict__ Zu1, const float* __restrict__ Zi1,
    const float* __restrict__ Eu, const float* __restrict__ Ei,
    const int* __restrict__ uids, const int* __restrict__ iids,
    const int* __restrict__ pos, const int* __restrict__ neg,
    const float* __restrict__ rs, const float* __restrict__ sumsq,
    float* __restrict__ out) {
  __shared__ float red[1024];
  int b = threadIdx.x;
  int u = uids[b], it = iids[b], p = pos[b], ng = neg[b];
  const float* EuU = Eu + (size_t)u * DIM;
  const float* EiI = Ei + (size_t)it * DIM;
  float p1 = clamp5(dot128(Gu + (size_t)u * DIM, EuU) / TEMPF);
  float p2 = clamp5(dot128(Gi + (size_t)it * DIM, EiI) / TEMPF);
  float p3 = clamp5(0.5f * dot128(Zu1 + (size_t)u * DIM, EuU) / TEMPF);
  float p4 = clamp5(0.5f * dot128(Zi1 + (size_t)it * DIM, EiI) / TEMPF);
  float n1 = logf(rs[0 * NB + b] + 1e-8f);
  float n2 = logf(rs[1 * NB + b] + 1e-8f);
  float n3 = logf(rs[2 * NB + b] + 1e-8f);
  float n4 = logf(rs[3 * NB + b] + 1e-8f);
  float dd = dot128(EuU, Ei + (size_t)p * DIM) - dot128(EuU, Ei + (size_t)ng * DIM);
  float lsig = fminf(dd, 0.0f) - log1pf(__expf(-fabsf(dd)));

  float sp1 = blockSum1024(p1, red), sp2 = blockSum1024(p2, red);
  float sp3 = blockSum1024(p3, red), sp4 = blockSum1024(p4, red);
  float sn1 = blockSum1024(n1, red), sn2 = blockSum1024(n2, red);
  float sn3 = blockSum1024(n3, red), sn4 = blockSum1024(n4, red);
  float sls = blockSum1024(lsig, red);

  if (b == 0) {
    const float inv = 1.0f / (float)NB;
    float pu = sp1 * inv, pi_ = sp2 * inv, pu2 = sp3 * inv, pi2 = sp4 * inv;
    float nu = sn1 * inv, ni_ = sn2 * inv, nu2 = sn3 * inv, ni2 = sn4 * inv;
    float loss_s = -(pu + pi_) + (nu + ni_);
    float loss_orig = -(pu2 + pi2) + (nu2 + ni2);
    float loss_cl = 0.5f * loss_orig + 0.5f * loss_s;
    float loss_r = -sls * inv;
    float loss_reg = 1e-7f * sumsq[0];
    float lcl = 0.2f * loss_cl;
    out[0] = loss_r + lcl + loss_reg;
    out[1] = loss_r;
    out[2] = lcl;
  }
}

extern "C" void kernel_launch(void* const* d_in, const int* in_sizes, int n_in,
                              void* d_out, int out_size, void* d_ws,
                              size_t ws_size, hipStream_t stream) {
  (void)in_sizes; (void)n_in; (void)out_size; (void)ws_size;
  const float* E_u_0   = (const float*)d_in[0];
  const float* E_i_0   = (const float*)d_in[1];
  const float* u_mul_s = (const float*)d_in[2];
  const float* v_mul_s = (const float*)d_in[3];
  const float* ut      = (const float*)d_in[4];
  const float* vt      = (const float*)d_in[5];
  const float* adj     = (const float*)d_in[6];
  const int* rows = (const int*)d_in[7];
  const int* cols = (const int*)d_in[8];
  const int* uids = (const int*)d_in[9];
  const int* iids = (const int*)d_in[10];
  const int* pos  = (const int*)d_in[11];
  const int* neg  = (const int*)d_in[12];

  const size_t ND = (size_t)N_USERS * DIM;  // 6,400,000 floats
  float* ws = (float*)d_ws;
  float* un   = ws;            float* inr = un + ND;
  float* uH   = inr + ND;      float* iH  = uH + ND;
  float* Zu1  = iH + ND;       float* Zi1 = Zu1 + ND;
  float* Zu2  = Zi1 + ND;      float* Zi2 = Zu2 + ND;
  float* Eu   = Zi2 + ND;      float* Ei  = Eu + ND;
  float* Gu   = Ei + ND;       float* Gi  = Gu + ND;
  float* simb = Gi + ND;       float* vals = simb + NE;
  float* rd   = vals + NE;     float* cd  = rd + N_USERS;
  float* Tu   = cd + N_ITEMS;  float* Ti  = Tu + QD * DIM;
  float* Anc  = Ti + QD * DIM;                  // 4 x NB x DIM
  float* rsum = Anc + 4 * (size_t)NB * DIM;     // 4 x NB
  float* ssq  = rsum + 4 * NB;                  // 1

  const int edgeBlocks = NE / 8;          // wave per edge, 8 waves/block
  const int normBlocks = (N_USERS * 32 + 255) / 256;

  hipMemsetAsync(Tu, 0, QD * DIM * sizeof(float), stream);
  hipMemsetAsync(Ti, 0, QD * DIM * sizeof(float), stream);
  hipMemsetAsync(rsum, 0, 4 * NB * sizeof(float), stream);
  hipMemsetAsync(ssq, 0, sizeof(float), stream);

  for (int l = 0; l < 2; l++) {
    const float* eu = (l == 0) ? E_u_0 : Zu1;
    const float* ei = (l == 0) ? E_i_0 : Zi1;
    float* ZuN = (l == 0) ? Zu1 : Zu2;
    float* ZiN = (l == 0) ? Zi1 : Zi2;

    normalize_rows_kernel<<<normBlocks, 256, 0, stream>>>(eu, un, N_USERS);
    normalize_rows_kernel<<<normBlocks, 256, 0, stream>>>(ei, inr, N_ITEMS);

    hipMemsetAsync(uH, 0, ND * sizeof(float), stream);
    hipMemsetAsync(iH, 0, ND * sizeof(float), stream);
    scatter_H_kernel<<<edgeBlocks, 256, 0, stream>>>(un, inr, rows, cols, adj,
                                                     uH, iH);
    hipMemsetAsync(rd, 0, N_USERS * sizeof(float), stream);
    hipMemsetAsync(cd, 0, N_ITEMS * sizeof(float), stream);
    edge_cos_kernel<<<edgeBlocks, 256, 0, stream>>>(uH, iH, rows, cols, simb,
                                                    rd, cd);
    edge_vals_kernel<<<(NE + 255) / 256, 256, 0, stream>>>(simb, rows, cols, rd,
                                                           cd, vals);
    hipMemsetAsync(ZuN, 0, ND * sizeof(float), stream);
    hipMemsetAsync(ZiN, 0, ND * sizeof(float), stream);
    scatter_Z_kernel<<<edgeBlocks, 256, 0, stream>>>(eu, ei, rows, cols, vals,
                                                     ZuN, ZiN);
    // Tu += vt @ ei ; Ti += ut @ eu   (32 tiles x 25 k-splits = 800 waves)
    gemm_qd_splitk_kernel<<<100, 256, 0, stream>>>(vt, ei, Tu, N_ITEMS, 2000);
    gemm_qd_splitk_kernel<<<100, 256, 0, stream>>>(ut, eu, Ti, N_USERS, 2000);
  }

  add3_kernel<<<(int)(ND / 256), 256, 0, stream>>>(E_u_0, Zu1, Zu2, Eu, (int)ND);
  add3_kernel<<<(int)(ND / 256), 256, 0, stream>>>(E_i_0, Zi1, Zi2, Ei, (int)ND);

  // G_u = E_u_0 + u_mul_s @ Tu ; G_i = E_i_0 + v_mul_s @ Ti
  gemm_md_kernel<<<(N_USERS / 16) * 8 / 8, 256, 0, stream>>>(u_mul_s, Tu, E_u_0,
                                                             Gu, N_USERS);
  gemm_md_kernel<<<(N_ITEMS / 16) * 8 / 8, 256, 0, stream>>>(v_mul_s, Ti, E_i_0,
                                                             Gi, N_ITEMS);

  float* A1 = Anc;
  float* A2 = Anc + (size_t)NB * DIM;
  float* A3 = Anc + 2 * (size_t)NB * DIM;
  float* A4 = Anc + 3 * (size_t)NB * DIM;
  gather_rows_kernel<<<NB, DIM, 0, stream>>>(Gu, uids, A1);
  gather_rows_kernel<<<NB, DIM, 0, stream>>>(Gi, iids, A2);
  gather_rows_kernel<<<NB, DIM, 0, stream>>>(Zu1, uids, A3);
  gather_rows_kernel<<<NB, DIM, 0, stream>>>(Zi1, iids, A4);

  // 64 anchor tiles x 3125 bank tiles = 200000 waves -> 25000 blocks
  const int infBlocks = (NB / 16) * (N_ITEMS / 16) / 8;
  infonce_neg_kernel<<<infBlocks, 256, 0, stream>>>(A1, Eu, 1.0f / TEMPF,
                                                    rsum + 0 * NB);
  infonce_neg_kernel<<<infBlocks, 256, 0, stream>>>(A2, Ei, 1.0f / TEMPF,
                                                    rsum + 1 * NB);
  infonce_neg_kernel<<<infBlocks, 256, 0, stream>>>(A3, Eu, 0.5f / TEMPF,
                                                    rsum + 2 * NB);
  infonce_neg_kernel<<<infBlocks, 256, 0, stream>>>(A4, Ei, 0.5f / TEMPF,
                                                    rsum + 3 * NB);

  sumsq_kernel<<<2048, 256, 0, stream>>>(E_u_0, (int)ND, ssq);
  sumsq_kernel<<<2048, 256, 0, stream>>>(E_i_0, (int)ND, ssq);

  final_kernel<<<1, 1024, 0, stream>>>(Gu, Gi, Zu1, Zi1, Eu, Ei, uids, iids,
                                       pos, neg, rsum, ssq, (float*)d_out);
}